// Model_73675868995847
// MI455X (gfx1250) — compile-verified
//
#include <hip/hip_runtime.h>
#include <math.h>

// Problem constants (from the reference)
#define B_  32
#define T_  512
#define H_  1024
#define NT_ 8
#define NE_ 24
#define C_  52           // C_TAGS + 2
#define START_ 50
#define STOP_  51
#define A_  36
#define D_  256

typedef __attribute__((ext_vector_type(2))) float v2f;
typedef __attribute__((ext_vector_type(4))) float v4f;
typedef __attribute__((ext_vector_type(8))) float v8f;

// ---------------------------------------------------------------------------
// fp32 WMMA GEMM: C[M,N] = A[M,K] @ B[K,N] (+ bias[N] if non-null)
// Requirements: M % 128 == 0, K % 128 == 0, N <= NTILES*16 (NTILES 3 or 4).
// Block = 128 threads (4 waves). Each wave owns TWO 16-row M-tiles and all
// NTILES 16-wide N-tiles. B is staged per 128-deep K-chunk into LDS in the
// V_WMMA_F32_16X16X4_F32 lane layout, with the NTILES float2 operands of a
// lane stored contiguously (stride 4) so they are fetched with ds_load_b128:
//   A (16x4 f32): lanes 0-15 -> M=lane, {K+0,K+1}; lanes 16-31 -> {K+2,K+3}
//   B (4x16 f32): lanes 0-15 -> N=lane, {K+0,K+1}; lanes 16-31 -> {K+2,K+3}
//   C/D: VGPR r: lanes 0-15 -> (M=r, N=lane); lanes 16-31 -> (M=8+r)
// Inner loop per k-group: 2 global_load_b64 (A) + <=2 ds_load_b128 (B)
// + 2*NTILES WMMAs. EXEC is all-1s throughout (no divergent branches).
// ---------------------------------------------------------------------------
template<int NTILES>
__global__ __launch_bounds__(128)
void gemm_f32_wmma(const float* __restrict__ A, const float* __restrict__ Bm,
                   const float* __restrict__ bias, float* __restrict__ C,
                   int M, int N, int K)
{
    constexpr int KC   = 128;        // K-chunk depth
    constexpr int NGRP = KC / 4;     // 32 k-groups per chunk
    __shared__ v2f smem[NGRP * 32 * 4];   // fixed 32 KB, nt-stride = 4

    const int tid  = threadIdx.x;
    const int wave = tid >> 5;
    const int lane = tid & 31;
    const int half = lane >> 4;      // 0: K+0/1, 1: K+2/3
    const int l    = lane & 15;

    const int mbase = blockIdx.x * 128 + wave * 32;   // two 16-row tiles

    const float* arow0 = A + (size_t)(mbase + l) * K + 2 * half;
    const float* arow1 = arow0 + (size_t)16 * K;

    v8f acc[2][NTILES];
    #pragma unroll
    for (int m = 0; m < 2; ++m)
        #pragma unroll
        for (int nt = 0; nt < NTILES; ++nt) acc[m][nt] = (v8f)(0.0f);

    for (int kbase = 0; kbase < K; kbase += KC) {
        __syncthreads();   // protect LDS against previous chunk's readers
        // ---- stage B chunk into LDS in WMMA lane layout (OOB cols -> 0) ----
        constexpr int ENTRIES = NGRP * NTILES * 32;   // logical float2 entries
        for (int e = tid; e < ENTRIES; e += 128) {
            int lane_e = e & 31;
            int nt_e   = (e >> 5) % NTILES;
            int g_e    = e / (32 * NTILES);
            int half_e = lane_e >> 4;
            int l_e    = lane_e & 15;
            int col    = nt_e * 16 + l_e;
            float msk  = (col < N) ? 1.0f : 0.0f;
            int   nc   = (col < N) ? col : (N - 1);
            int   row  = kbase + 4 * g_e + 2 * half_e;
            v2f v;
            v.x = Bm[(size_t)row * N + nc] * msk;
            v.y = Bm[(size_t)(row + 1) * N + nc] * msk;
            smem[(g_e * 32 + lane_e) * 4 + nt_e] = v;
        }
        __syncthreads();

        // ---- compute over this K-chunk ----
        for (int g = 0; g < NGRP; ++g) {
            v2f a0 = *(const v2f*)(arow0 + kbase + 4 * g);
            v2f a1 = *(const v2f*)(arow1 + kbase + 4 * g);

            const v2f* bptr = smem + (g * 32 + lane) * 4;   // 32B aligned
            v2f b[NTILES];
            {
                v4f b01 = *(const v4f*)bptr;                // ds_load_b128
                b[0] = __builtin_shufflevector(b01, b01, 0, 1);
                b[1] = __builtin_shufflevector(b01, b01, 2, 3);
            }
            if constexpr (NTILES == 3) {
                b[2] = bptr[2];                             // ds_load_b64
            } else if constexpr (NTILES == 4) {
                v4f b23 = *(const v4f*)(bptr + 2);          // ds_load_b128
                b[2] = __builtin_shufflevector(b23, b23, 0, 1);
                b[3] = __builtin_shufflevector(b23, b23, 2, 3);
            }

            #pragma unroll
            for (int nt = 0; nt < NTILES; ++nt) {
                acc[0][nt] = __builtin_amdgcn_wmma_f32_16x16x4_f32(
                    false, a0, false, b[nt], (short)0, acc[0][nt], false, false);
                acc[1][nt] = __builtin_amdgcn_wmma_f32_16x16x4_f32(
                    false, a1, false, b[nt], (short)0, acc[1][nt], false, false);
            }
        }
    }

    // ---- store (bias fused) ----
    #pragma unroll
    for (int m = 0; m < 2; ++m) {
        int mt = mbase + m * 16;
        #pragma unroll
        for (int nt = 0; nt < NTILES; ++nt) {
            int col = nt * 16 + l;
            if (col < N) {
                float bv = bias ? bias[col] : 0.0f;
                #pragma unroll
                for (int r = 0; r < 8; ++r)
                    C[(size_t)(mt + half * 8 + r) * N + col] = acc[m][nt][r] + bv;
            }
        }
    }
}

// ---------------------------------------------------------------------------
// out[n] = b2[n] + sum_d b1[d] * W2[d, n]
// ---------------------------------------------------------------------------
__global__ void fold_bias(const float* __restrict__ b1, const float* __restrict__ W2,
                          const float* __restrict__ b2, float* __restrict__ out,
                          int Din, int Nout)
{
    int n = blockIdx.x * blockDim.x + threadIdx.x;
    if (n >= Nout) return;
    float s = b2[n];
    for (int d = 0; d < Din; ++d) s += b1[d] * W2[d * Nout + n];
    out[n] = s;
}

// ---------------------------------------------------------------------------
// Span mean: out[b, i, h] = mean_{t in [s, e)} enc[b, t, h]
// ---------------------------------------------------------------------------
__global__ void span_mean_kernel(const float* __restrict__ enc,
                                 const int* __restrict__ starts,
                                 const int* __restrict__ ends,
                                 float* __restrict__ out, int NS)
{
    int bi = blockIdx.x;             // b*NS + i
    int b  = bi / NS;
    int s  = starts[bi];
    int e  = ends[bi];
    int len = e - s;
    float inv = 1.0f / (float)len;
    const float* base = enc + ((size_t)b * T_ + s) * H_;
    for (int h = threadIdx.x; h < H_; h += blockDim.x) {
        float acc = 0.0f;
        for (int t = 0; t < len; ++t) acc += base[(size_t)t * H_ + h];
        out[(size_t)bi * H_ + h] = acc * inv;
    }
}

// ---------------------------------------------------------------------------
// logits[b,t,e,a] = (Tr[b,t,a] + E[b,e,a] + ab[a]) * pair_mask
// ---------------------------------------------------------------------------
__global__ void combine_logits(const float* __restrict__ Tr, const float* __restrict__ E,
                               const float* __restrict__ ab,
                               const int* __restrict__ trig_mask,
                               const int* __restrict__ ent_mask,
                               float* __restrict__ out)
{
    int idx = blockIdx.x * blockDim.x + threadIdx.x;
    const int total = B_ * NT_ * NE_ * A_;
    if (idx >= total) return;
    int a = idx % A_;
    int e = (idx / A_) % NE_;
    int t = (idx / (A_ * NE_)) % NT_;
    int b = idx / (A_ * NE_ * NT_);
    float v = Tr[(b * NT_ + t) * A_ + a] + E[(b * NE_ + e) * A_ + a] + ab[a];
    float m = (trig_mask[b * NT_ + t] && ent_mask[b * NE_ + e]) ? 1.0f : 0.0f;
    out[idx] = v * m;
}

// ---------------------------------------------------------------------------
// CRF negative log-likelihood. One block (64 threads) per batch element.
// Mask t<len is monotone -> scan runs to lengths[b].
// ---------------------------------------------------------------------------
__global__ __launch_bounds__(64)
void crf_nll_kernel(const float* __restrict__ feats, const int* __restrict__ tags,
                    const int* __restrict__ lengths, const float* __restrict__ trans,
                    float* __restrict__ loss)
{
    __shared__ float tr[C_ * C_];
    __shared__ float alpha[C_];
    __shared__ float red[C_];
    __shared__ float gpart[64];

    const int b   = blockIdx.x;
    const int tid = threadIdx.x;

    for (int i = tid; i < C_ * C_; i += 64) tr[i] = trans[i];
    __syncthreads();

    const float* fb = feats + (size_t)b * T_ * C_;
    const int*   tb = tags + b * T_;
    const int   len = lengths[b];

    if (tid < C_) alpha[tid] = fb[tid] + tr[START_ * C_ + tid];
    __syncthreads();

    for (int t = 1; t < len; ++t) {
        float nv = 0.0f;
        if (tid < C_) {
            float m = -INFINITY;
            for (int i = 0; i < C_; ++i)
                m = fmaxf(m, alpha[i] + tr[i * C_ + tid]);
            float s = 0.0f;
            for (int i = 0; i < C_; ++i)
                s += __expf(alpha[i] + tr[i * C_ + tid] - m);
            nv = m + __logf(s) + fb[(size_t)t * C_ + tid];
        }
        __syncthreads();
        if (tid < C_) alpha[tid] = nv;
        __syncthreads();
    }

    if (tid < C_) red[tid] = alpha[tid] + tr[tid * C_ + STOP_];

    float part = 0.0f;
    for (int t = tid; t < len; t += 64) {
        part += fb[(size_t)t * C_ + tb[t]];
        if (t >= 1) part += tr[tb[t - 1] * C_ + tb[t]];
    }
    gpart[tid] = part;
    __syncthreads();

    if (tid == 0) {
        float m = red[0];
        for (int j = 1; j < C_; ++j) m = fmaxf(m, red[j]);
        float s = 0.0f;
        for (int j = 0; j < C_; ++j) s += __expf(red[j] - m);
        float logZ = m + __logf(s);
        float gold = tr[START_ * C_ + tb[0]] + tr[tb[len - 1] * C_ + STOP_];
        for (int i = 0; i < 64; ++i) gold += gpart[i];
        atomicAdd(loss, logZ - gold);
    }
}

__global__ void zero_scalar(float* p) { p[0] = 0.0f; }

// ---------------------------------------------------------------------------
extern "C" void kernel_launch(void* const* d_in, const int* in_sizes, int n_in,
                              void* d_out, int out_size, void* d_ws, size_t ws_size,
                              hipStream_t stream)
{
    const float* enc     = (const float*)d_in[0];   // (B,T,H)
    const float* fc_W1   = (const float*)d_in[1];   // (H,D)
    const float* fc_b1   = (const float*)d_in[2];   // (D)
    const float* fc_W2   = (const float*)d_in[3];   // (D,C)
    const float* fc_b2   = (const float*)d_in[4];   // (C)
    const float* trans   = (const float*)d_in[5];   // (C,C)
    const float* arg_W1  = (const float*)d_in[6];   // (2H,D)
    const float* arg_b1  = (const float*)d_in[7];   // (D)
    const float* arg_W2  = (const float*)d_in[8];   // (D,A)
    const float* arg_b2  = (const float*)d_in[9];   // (A)
    const int*   tags    = (const int*)d_in[10];    // (B,T)
    const int*   lengths = (const int*)d_in[11];    // (B)
    const int*   trig_s  = (const int*)d_in[12];    // (B,NT)
    const int*   trig_e  = (const int*)d_in[13];
    const int*   trig_m  = (const int*)d_in[14];
    const int*   ent_s   = (const int*)d_in[15];    // (B,NE)
    const int*   ent_e   = (const int*)d_in[16];
    const int*   ent_m   = (const int*)d_in[17];

    float* out_loss   = (float*)d_out;              // [0]
    float* out_logits = (float*)d_out + 1;          // (B*NT*NE, A)

    // workspace carve-up (floats); total ~8.3 MB
    float* ws      = (float*)d_ws;
    float* W12     = ws;                             // H x C
    float* b12     = W12 + (size_t)H_ * C_;          // 64
    float* feats   = b12 + 64;                       // (B*T) x C
    float* argW12  = feats + (size_t)B_ * T_ * C_;   // 2H x A
    float* ab      = argW12 + (size_t)2 * H_ * A_;   // 64
    float* tvec    = ab + 64;                        // (B*NT) x H
    float* evec    = tvec + (size_t)B_ * NT_ * H_;   // (B*NE) x H
    float* Tr      = evec + (size_t)B_ * NE_ * H_;   // (B*NT) x A
    float* E       = Tr + (size_t)B_ * NT_ * A_;     // (B*NE) x A

    // 0) loss accumulator = 0
    zero_scalar<<<1, 1, 0, stream>>>(out_loss);

    // 1) W12 = fc_W1 @ fc_W2   (M=1024, N=52, K=256)
    gemm_f32_wmma<4><<<(H_ / 128), 128, 0, stream>>>(fc_W1, fc_W2, nullptr, W12,
                                                     H_, C_, D_);
    // 2) b12 = fc_b1 @ fc_W2 + fc_b2
    fold_bias<<<1, 64, 0, stream>>>(fc_b1, fc_W2, fc_b2, b12, D_, C_);

    // 3) feats = enc @ W12 + b12   (M=16384, N=52, K=1024)
    gemm_f32_wmma<4><<<(B_ * T_ / 128), 128, 0, stream>>>(enc, W12, b12, feats,
                                                          B_ * T_, C_, H_);

    // 4) CRF loss (depends on feats)
    crf_nll_kernel<<<B_, 64, 0, stream>>>(feats, tags, lengths, trans, out_loss);

    // 5) argW12 = arg_W1 @ arg_W2   (M=2048, N=36, K=256)
    gemm_f32_wmma<3><<<(2 * H_ / 128), 128, 0, stream>>>(arg_W1, arg_W2, nullptr,
                                                         argW12, 2 * H_, A_, D_);
    // 6) ab = arg_b1 @ arg_W2 + arg_b2
    fold_bias<<<1, 64, 0, stream>>>(arg_b1, arg_W2, arg_b2, ab, D_, A_);

    // 7) span means (direct sums, spans <= 16)
    span_mean_kernel<<<B_ * NT_, 256, 0, stream>>>(enc, trig_s, trig_e, tvec, NT_);
    span_mean_kernel<<<B_ * NE_, 256, 0, stream>>>(enc, ent_s, ent_e, evec, NE_);

    // 8) Tr = trig_vec @ Whi (M=256); E = ent_vec @ Wlo (M=768); N=36, K=1024
    gemm_f32_wmma<3><<<(B_ * NT_ / 128), 128, 0, stream>>>(
        tvec, argW12 + (size_t)H_ * A_, nullptr, Tr, B_ * NT_, A_, H_);
    gemm_f32_wmma<3><<<(B_ * NE_ / 128), 128, 0, stream>>>(
        evec, argW12, nullptr, E, B_ * NE_, A_, H_);

    // 9) logits = (Tr + E + ab) * pair_mask
    {
        const int total = B_ * NT_ * NE_ * A_;
        combine_logits<<<(total + 255) / 256, 256, 0, stream>>>(
            Tr, E, ab, trig_m, ent_m, out_logits);
    }
}